// GIN_42795054137779
// MI455X (gfx1250) — compile-verified
//
#include <hip/hip_runtime.h>
#include <hip/hip_bf16.h>

// ---------------------------------------------------------------------------
// GIN forward on MI455X (gfx1250).
// GEMMs use V_WMMA_F32_16X16X4_F32 (exact f32 matrix path, wave32), with
// weights staged in LDS (320 KB/WGP) and 4 independent accumulators per wave
// so WMMAs issue back-to-back without D->C hazard stalls on every op.
// Workload is HBM-bound (~2 GB traffic, ~5.7 GFLOP) => f32 WMMA is optimal:
// exact vs the f32 reference, matrix pipe nowhere near the bottleneck.
// ---------------------------------------------------------------------------

typedef float v2f __attribute__((ext_vector_type(2)));
typedef float v8f __attribute__((ext_vector_type(8)));

#define N_NODES 100000
#define N_FEAT  128
#define HID     64
#define N_GRAPHS 512
#define N_LAYERS 3
#define POOL_W  (HID * N_LAYERS)   // 192
#define K_MAX   192                // largest GEMM K (classifier)
#define BN_EPS  1e-5f

// ---------------------------------------------------------------------------
// Utility: zero floats
// ---------------------------------------------------------------------------
__global__ void zero_f32(float* __restrict__ p, size_t n) {
  size_t i = (size_t)blockIdx.x * blockDim.x + threadIdx.x;
  if (i < n) p[i] = 0.0f;
}

// Copy (float4 vectorized); n4 = count of float4s
__global__ void copy_f32x4(const float4* __restrict__ src, float4* __restrict__ dst, size_t n4) {
  size_t i = (size_t)blockIdx.x * blockDim.x + threadIdx.x;
  if (i < n4) dst[i] = src[i];
}

// ---------------------------------------------------------------------------
// Edge scatter-add: AGG[dst[e]][:] += H[src[e]][:]   (AGG pre-initialized = H,
// fusing the "h + agg" step). One thread per (edge, 4-feature chunk).
// ---------------------------------------------------------------------------
__global__ void scatter_edges(const int* __restrict__ esrc, const int* __restrict__ edst,
                              const float* __restrict__ H, float* __restrict__ AGG,
                              int E, int d) {
  const int d4 = d >> 2;
  size_t idx = (size_t)blockIdx.x * blockDim.x + threadIdx.x;
  size_t total = (size_t)E * d4;
  if (idx >= total) return;
  int e = (int)(idx / d4);
  int f = ((int)(idx - (size_t)e * d4)) << 2;
  const float4 v = *(const float4*)(H + (size_t)esrc[e] * d + f);
  float* out = AGG + (size_t)edst[e] * d + f;
  atomicAdd(out + 0, v.x);
  atomicAdd(out + 1, v.y);
  atomicAdd(out + 2, v.z);
  atomicAdd(out + 3, v.w);
}

// ---------------------------------------------------------------------------
// WMMA GEMM: OUT[M x 64] = IN[M x K] @ W[K x 64] + bias  (optional ReLU).
// Row-major everywhere.
//
// Block = 256 threads = 8 wave32s. W (K x 64, <= 48 KB) is staged into LDS
// once per block. Each wave owns a 16-row strip and all 64 output columns:
// per k-step it loads one A fragment (b64) from global and 4 B fragments
// from LDS, issuing 4 independent v_wmma_f32_16x16x4_f32 (4 accumulators).
// Block covers 128 rows.
//
// f32 operand layouts (ISA 7.12.2):
//   A 16x4 : lane half h=lane>>4 selects K pair {2h, 2h+1}; row = lane&15.
//   B 4x16 : same K-pair split across lane halves; col = lane&15.
//   C/D    : VGPR v holds row (v + 8h), col = lane&15.
// ---------------------------------------------------------------------------
__global__ __launch_bounds__(256)
void gemm_bias_wmma(const float* __restrict__ IN, const float* __restrict__ W,
                    const float* __restrict__ bias, float* __restrict__ OUT,
                    int M, int K, int do_relu) {
  __shared__ float sW[K_MAX * HID];      // 48 KB max (LDS is 320 KB/WGP)

  const int tid  = threadIdx.x;
  const int lane = tid & 31;
  const int wave = tid >> 5;             // 0..7
  const int half = lane >> 4;            // 0 or 1
  const int l15  = lane & 15;

  // Cooperative W -> LDS stage (float4 = b128 loads)
  {
    const int n4 = K * (HID / 4);
    for (int i = tid; i < n4; i += 256)
      ((float4*)sW)[i] = ((const float4*)W)[i];
  }
  __syncthreads();

  const int row0 = (blockIdx.x * 8 + wave) * 16;
  const int arow = row0 + l15;
  // Clamp row for loads so EXEC stays all-ones through the WMMA loop; rows
  // beyond M produce garbage that is masked at the store.
  const int arow_c = (arow < M) ? arow : (M - 1);
  const float* aptr = IN + (size_t)arow_c * K + (half << 1);

  v8f acc[4] = {};
#pragma unroll 2
  for (int k0 = 0; k0 < K; k0 += 4) {
    v2f a;
    a.x = aptr[k0 + 0];
    a.y = aptr[k0 + 1];
    const int kb = (k0 + (half << 1)) * HID + l15;
#pragma unroll
    for (int nt = 0; nt < 4; ++nt) {
      v2f b;
      b.x = sW[kb + nt * 16];
      b.y = sW[kb + HID + nt * 16];
      // (neg_a, A, neg_b, B, c_mod, C, reuse_a, reuse_b)
      acc[nt] = __builtin_amdgcn_wmma_f32_16x16x4_f32(false, a, false, b,
                                                      (short)0, acc[nt], false, false);
    }
  }

#pragma unroll
  for (int nt = 0; nt < 4; ++nt) {
    const int col = nt * 16 + l15;
    const float bv = bias[col];
#pragma unroll
    for (int v = 0; v < 8; ++v) {
      int r = row0 + v + (half << 3);
      if (r < M) {
        float o = acc[nt][v] + bv;
        if (do_relu) o = (o > 0.0f) ? o : 0.0f;
        OUT[(size_t)r * HID + col] = o;
      }
    }
  }
}

// ---------------------------------------------------------------------------
// BatchNorm pass 1: per-feature sum and sum-of-squares over M rows (64 feats).
// stats[0..63] = sum, stats[64..127] = sumsq. Must be zeroed beforehand.
// ---------------------------------------------------------------------------
__global__ __launch_bounds__(256)
void bn_stats(const float* __restrict__ H, int M, float* __restrict__ stats) {
  __shared__ float s_sum[256];
  __shared__ float s_sq[256];
  const int f    = threadIdx.x & 63;
  const int rgrp = threadIdx.x >> 6;      // 0..3
  float lsum = 0.0f, lsq = 0.0f;
  for (int r = blockIdx.x * 4 + rgrp; r < M; r += gridDim.x * 4) {
    float v = H[(size_t)r * HID + f];
    lsum += v;
    lsq  += v * v;
  }
  s_sum[threadIdx.x] = lsum;
  s_sq[threadIdx.x]  = lsq;
  __syncthreads();
  if (threadIdx.x < 64) {
    float ts = s_sum[f] + s_sum[f + 64] + s_sum[f + 128] + s_sum[f + 192];
    float tq = s_sq[f]  + s_sq[f + 64]  + s_sq[f + 128]  + s_sq[f + 192];
    atomicAdd(&stats[f], ts);
    atomicAdd(&stats[64 + f], tq);
  }
}

// BatchNorm pass 2 (+ReLU), in place, float4-vectorized (16 threads per row).
// Biased variance (matches jnp.var).
__global__ void bn_apply_relu(float* __restrict__ H, const float* __restrict__ stats,
                              const float* __restrict__ g, const float* __restrict__ be,
                              int M) {
  size_t idx = (size_t)blockIdx.x * blockDim.x + threadIdx.x;   // one float4 each
  size_t total = (size_t)M * (HID / 4);
  if (idx >= total) return;
  const int f0 = ((int)(idx & 15)) << 2;    // feature base 0,4,...,60
  const float invM = 1.0f / (float)M;
  float4 v = ((float4*)H)[idx];
  float sc[4], sh[4];
#pragma unroll
  for (int j = 0; j < 4; ++j) {
    int f = f0 + j;
    float mean = stats[f] * invM;
    float var  = stats[64 + f] * invM - mean * mean;
    sc[j] = g[f] * rsqrtf(var + BN_EPS);
    sh[j] = be[f] - mean * sc[j];
  }
  v.x = fmaxf(v.x * sc[0] + sh[0], 0.0f);
  v.y = fmaxf(v.y * sc[1] + sh[1], 0.0f);
  v.z = fmaxf(v.z * sc[2] + sh[2], 0.0f);
  v.w = fmaxf(v.w * sc[3] + sh[3], 0.0f);
  ((float4*)H)[idx] = v;
}

// ---------------------------------------------------------------------------
// Graph pooling: pooled[batch[i]][layerOff + f] += H[i][f]
// ---------------------------------------------------------------------------
__global__ void pool_scatter(const float* __restrict__ H, const int* __restrict__ batch,
                             float* __restrict__ pooled, int M, int layerOff) {
  size_t idx = (size_t)blockIdx.x * blockDim.x + threadIdx.x;
  size_t total = (size_t)M * HID;
  if (idx >= total) return;
  int i = (int)(idx >> 6);
  int f = (int)(idx & 63);
  atomicAdd(&pooled[(size_t)batch[i] * POOL_W + layerOff + f], H[idx]);
}

// Final tiny head: out[g][c] = Z[g][:] . Wc2[:,c] + bc2[c]   (512 x 2)
__global__ void clf_out(const float* __restrict__ Z, const float* __restrict__ Wc2,
                        const float* __restrict__ bc2, float* __restrict__ out,
                        int G, int C) {
  int idx = blockIdx.x * blockDim.x + threadIdx.x;
  if (idx >= G * C) return;
  int g = idx / C;
  int c = idx - g * C;
  float acc = bc2[c];
#pragma unroll 8
  for (int k = 0; k < HID; ++k) acc += Z[(size_t)g * HID + k] * Wc2[k * C + c];
  out[idx] = acc;
}

// ---------------------------------------------------------------------------
// Launch
// ---------------------------------------------------------------------------
static inline int nblk(size_t n, int b) { return (int)((n + b - 1) / b); }

extern "C" void kernel_launch(void* const* d_in, const int* in_sizes, int n_in,
                              void* d_out, int out_size, void* d_ws, size_t ws_size,
                              hipStream_t stream) {
  const float* x    = (const float*)d_in[0];
  const int*   eidx = (const int*)d_in[1];
  const int*   batch = (const int*)d_in[2];
  const int E = in_sizes[1] / 2;
  const int M = in_sizes[0] / N_FEAT;     // 100000
  const int* esrc = eidx;                 // edge_index[0]
  const int* edst = eidx + E;             // edge_index[1]

  // Per-layer params start at index 4; insertion order:
  // W1, b1, g1, be1, W2, b2, g_out, be_out
  const float* Wc1 = (const float*)d_in[4 + 8 * N_LAYERS + 0];
  const float* bc1 = (const float*)d_in[4 + 8 * N_LAYERS + 1];
  const float* Wc2 = (const float*)d_in[4 + 8 * N_LAYERS + 2];
  const float* bc2 = (const float*)d_in[4 + 8 * N_LAYERS + 3];

  // Workspace layout (floats)
  float* ws = (float*)d_ws;
  float* hagg   = ws;                                  // 100000 x 128 max
  float* hmid   = hagg + (size_t)N_NODES * N_FEAT;     // 100000 x 64
  float* hcur   = hmid + (size_t)N_NODES * HID;        // 100000 x 64
  float* pooled = hcur + (size_t)N_NODES * HID;        // 512 x 192
  float* zbuf   = pooled + (size_t)N_GRAPHS * POOL_W;  // 512 x 64
  float* stats  = zbuf + (size_t)N_GRAPHS * HID;       // 128

  const int T = 256;
  const int GEMM_ROWS = 128;              // rows per block (8 waves x 16)

  // Zero pooled accumulator
  zero_f32<<<nblk((size_t)N_GRAPHS * POOL_W, T), T, 0, stream>>>(
      pooled, (size_t)N_GRAPHS * POOL_W);

  const float* hsrc = x;
  for (int l = 0; l < N_LAYERS; ++l) {
    const int din = (l == 0) ? N_FEAT : HID;
    const float* const* P = (const float* const*)(d_in + 4 + 8 * l);
    const float* W1 = P[0]; const float* b1  = P[1];
    const float* g1 = P[2]; const float* be1 = P[3];
    const float* W2 = P[4]; const float* b2  = P[5];
    const float* go = P[6]; const float* beo = P[7];

    // hagg = hsrc  (fuses "h + agg")
    size_t n4 = (size_t)M * din / 4;
    copy_f32x4<<<nblk(n4, T), T, 0, stream>>>((const float4*)hsrc, (float4*)hagg, n4);

    // hagg[dst] += hsrc[src]
    size_t chunks = (size_t)E * (din >> 2);
    scatter_edges<<<nblk(chunks, T), T, 0, stream>>>(esrc, edst, hsrc, hagg, E, din);

    // hmid = hagg @ W1 + b1
    gemm_bias_wmma<<<nblk((size_t)M, GEMM_ROWS), T, 0, stream>>>(hagg, W1, b1, hmid, M, din, 0);

    // BN1 + ReLU
    zero_f32<<<1, 128, 0, stream>>>(stats, 128);
    bn_stats<<<1024, T, 0, stream>>>(hmid, M, stats);
    bn_apply_relu<<<nblk((size_t)M * (HID / 4), T), T, 0, stream>>>(hmid, stats, g1, be1, M);

    // hcur = hmid @ W2 + b2
    gemm_bias_wmma<<<nblk((size_t)M, GEMM_ROWS), T, 0, stream>>>(hmid, W2, b2, hcur, M, HID, 0);

    // BN_out + ReLU
    zero_f32<<<1, 128, 0, stream>>>(stats, 128);
    bn_stats<<<1024, T, 0, stream>>>(hcur, M, stats);
    bn_apply_relu<<<nblk((size_t)M * (HID / 4), T), T, 0, stream>>>(hcur, stats, go, beo, M);

    // pooled[:, l*64:(l+1)*64] += segment_sum(hcur, batch)
    pool_scatter<<<nblk((size_t)M * HID, T), T, 0, stream>>>(hcur, batch, pooled, M, l * HID);

    hsrc = hcur;
  }

  // z = relu(pooled @ Wc1 + bc1)   [512 x 192] @ [192 x 64]
  gemm_bias_wmma<<<nblk((size_t)N_GRAPHS, GEMM_ROWS), T, 0, stream>>>(
      pooled, Wc1, bc1, zbuf, N_GRAPHS, POOL_W, 1);

  // out = z @ Wc2 + bc2            [512 x 64] @ [64 x 2]
  clf_out<<<nblk((size_t)N_GRAPHS * 2, T), T, 0, stream>>>(
      zbuf, Wc2, bc2, (float*)d_out, N_GRAPHS, 2);
}